// SetAbstraction_48421461295499
// MI455X (gfx1250) — compile-verified
//
#include <hip/hip_runtime.h>
#include <hip/hip_bf16.h>

typedef __attribute__((ext_vector_type(2))) float v2f;
typedef __attribute__((ext_vector_type(8))) float v8f;

#define NCOL 8192      // B * NP = 8 * 1024 columns
#define BN_EPS 1e-5f

// ---------------------------------------------------------------------------
// Kernel 1: grouped_xyz output is exactly zero (cxt - cxt). 1,572,864 floats.
// ---------------------------------------------------------------------------
__global__ void k_zero(float4* __restrict__ p) {
    int i = blockIdx.x * blockDim.x + threadIdx.x;   // 1536*256 = 393216 float4
    p[i] = make_float4(0.f, 0.f, 0.f, 0.f);
}

// ---------------------------------------------------------------------------
// Kernel 2: gather 9-dim feature per column and apply layer-1 GEMM (64x9).
// Y1 laid out (64, NCOL) so later column accesses are coalesced.
// ---------------------------------------------------------------------------
__global__ void k_gather_l1(const float* __restrict__ points,   // (8,6,4096)
                            const float* __restrict__ onehot,   // (8,3)
                            const int*   __restrict__ idx,      // (8,1024)
                            const float* __restrict__ w1,       // (64,9)
                            float* __restrict__ Y1)             // (64, NCOL)
{
    int j = blockIdx.x * blockDim.x + threadIdx.x;   // 0..8191
    int b = j >> 10;
    int p = j & 1023;
    int i = idx[b * 1024 + p];

    float x[9];
#pragma unroll
    for (int c = 0; c < 6; ++c)
        x[c] = points[((size_t)b * 6 + c) * 4096 + i];
#pragma unroll
    for (int c = 0; c < 3; ++c)
        x[6 + c] = onehot[b * 3 + c];

#pragma unroll 4
    for (int o = 0; o < 64; ++o) {
        float acc = 0.f;
#pragma unroll
        for (int c = 0; c < 9; ++c)
            acc = fmaf(w1[o * 9 + c], x[c], acc);
        Y1[(size_t)o * NCOL + j] = acc;
    }
}

// ---------------------------------------------------------------------------
// Kernel 3: per-channel sum / sum-of-squares (deterministic LDS reduction,
// one block per channel). stats[2*ch] = sum, stats[2*ch+1] = sumsq.
// ---------------------------------------------------------------------------
__global__ void k_stats(const float* __restrict__ Y, float* __restrict__ stats) {
    int ch = blockIdx.x;
    const float* row = Y + (size_t)ch * NCOL;
    float s = 0.f, ss = 0.f;
    for (int j = threadIdx.x; j < NCOL; j += 256) {
        float v = row[j];
        s += v;
        ss = fmaf(v, v, ss);
    }
    __shared__ float sh_s[256];
    __shared__ float sh_ss[256];
    sh_s[threadIdx.x] = s;
    sh_ss[threadIdx.x] = ss;
    __syncthreads();
#pragma unroll
    for (int off = 128; off > 0; off >>= 1) {
        if (threadIdx.x < off) {
            sh_s[threadIdx.x]  += sh_s[threadIdx.x + off];
            sh_ss[threadIdx.x] += sh_ss[threadIdx.x + off];
        }
        __syncthreads();
    }
    if (threadIdx.x == 0) {
        stats[2 * ch]     = sh_s[0];
        stats[2 * ch + 1] = sh_ss[0];
    }
}

// ---------------------------------------------------------------------------
// Kernel 4: fused BN(prev)+ReLU -> GEMM via V_WMMA_F32_16X16X4_F32.
// Yout(Cout, NCOL) = W(Cout, Cin) * relu(bn(Yprev(Cin, NCOL))).
// One wave computes a 16(M) x 16(N) f32 tile; K marched 4 at a time.
//
// ISA fragment layouts (wave32):
//   A 16x4:  lanes 0-15 -> {K=k0, k0+1}, lanes 16-31 -> {K=k0+2, k0+3}
//   B 4x16:  VGPR0 holds rows K (lanes 0-15) / K+2 (lanes 16-31), VGPR1 K+1/K+3
//   C/D:     VGPR r holds row mBase+r (lanes 0-15) and mBase+r+8 (lanes 16-31)
// ---------------------------------------------------------------------------
__global__ void k_gemm_bn_relu_wmma(const float* __restrict__ Yprev,
                                    const float* __restrict__ stats,
                                    const float* __restrict__ gamma,
                                    const float* __restrict__ beta,
                                    const float* __restrict__ W,
                                    float* __restrict__ Yout,
                                    int Cin, int Cout)
{
    __shared__ float s_scale[64];
    __shared__ float s_shift[64];
    int t = threadIdx.x;
    if (t < Cin) {
        float sum = stats[2 * t], sumsq = stats[2 * t + 1];
        float mu  = sum * (1.0f / NCOL);
        float var = fmaf(-mu, mu, sumsq * (1.0f / NCOL));
        float rs  = rsqrtf(var + BN_EPS);
        float sc  = gamma[t] * rs;
        s_scale[t] = sc;
        s_shift[t] = fmaf(-mu, sc, beta[t]);
    }
    __syncthreads();

    int wave = t >> 5;
    int lane = t & 31;
    int lh   = lane >> 4;    // which half of the wave
    int lm   = lane & 15;

    int ntilesM = Cout >> 4;
    int tiles   = ntilesM * (NCOL / 16);
    int tile    = blockIdx.x * 8 + wave;     // wave-uniform
    if (tile >= tiles) return;
    int mt = tile % ntilesM;
    int nt = tile / ntilesM;
    int mBase = mt * 16, nBase = nt * 16;

    int mRow = mBase + lm;                   // A row this lane supplies
    int n    = nBase + lm;                   // B/D column this lane supplies

    v8f acc = {};
    for (int k0 = 0; k0 < Cin; k0 += 4) {
        int ka = k0 + 2 * lh;
        v2f a, bf;
        a.x = W[(size_t)mRow * Cin + ka];
        a.y = W[(size_t)mRow * Cin + ka + 1];
        float r0 = Yprev[(size_t)ka * NCOL + n];
        float r1 = Yprev[(size_t)(ka + 1) * NCOL + n];
        r0 = fmaf(r0, s_scale[ka],     s_shift[ka]);
        r1 = fmaf(r1, s_scale[ka + 1], s_shift[ka + 1]);
        bf.x = fmaxf(r0, 0.f);
        bf.y = fmaxf(r1, 0.f);
        acc = __builtin_amdgcn_wmma_f32_16x16x4_f32(
            /*neg_a=*/false, a, /*neg_b=*/false, bf,
            /*c_mod=*/(short)0, acc, /*reuse_a=*/false, /*reuse_b=*/false);
    }

#pragma unroll
    for (int r = 0; r < 8; ++r) {
        int m = mBase + r + 8 * lh;
        Yout[(size_t)m * NCOL + n] = acc[r];
    }
}

// ---------------------------------------------------------------------------
// Kernel 5: BN+ReLU of Y3 and scatter to pooled (B, 128, NP) layout.
// ---------------------------------------------------------------------------
__global__ void k_finalize(const float* __restrict__ Y3,     // (128, NCOL)
                           const float* __restrict__ stats,
                           const float* __restrict__ gamma,
                           const float* __restrict__ beta,
                           float* __restrict__ pooled)       // (8, 128, 1024)
{
    int id = blockIdx.x * blockDim.x + threadIdx.x;  // 128 * NCOL
    int o  = id >> 13;       // channel
    int j  = id & (NCOL - 1);
    int b  = j >> 10;
    int p  = j & 1023;

    float sum = stats[2 * o], sumsq = stats[2 * o + 1];
    float mu  = sum * (1.0f / NCOL);
    float var = fmaf(-mu, mu, sumsq * (1.0f / NCOL));
    float rs  = rsqrtf(var + BN_EPS);
    float sc  = gamma[o] * rs;
    float sh  = fmaf(-mu, sc, beta[o]);

    float v = fmaxf(fmaf(Y3[id], sc, sh), 0.f);
    pooled[(size_t)b * (128 * 1024) + o * 1024 + p] = v;
}

// ---------------------------------------------------------------------------
extern "C" void kernel_launch(void* const* d_in, const int* in_sizes, int n_in,
                              void* d_out, int out_size, void* d_ws, size_t ws_size,
                              hipStream_t stream) {
    // setup_inputs order:
    // 0 xyz, 1 points, 2 one_hot_vec, 3 idx, 4 w1, 5 g1, 6 b1,
    // 7 w2, 8 g2, 9 b2, 10 w3, 11 g3, 12 b3
    const float* points = (const float*)d_in[1];
    const float* onehot = (const float*)d_in[2];
    const int*   idx    = (const int*)  d_in[3];
    const float* w1 = (const float*)d_in[4];
    const float* g1 = (const float*)d_in[5];
    const float* b1 = (const float*)d_in[6];
    const float* w2 = (const float*)d_in[7];
    const float* g2 = (const float*)d_in[8];
    const float* b2 = (const float*)d_in[9];
    const float* w3 = (const float*)d_in[10];
    const float* g3 = (const float*)d_in[11];
    const float* b3 = (const float*)d_in[12];

    float* out = (float*)d_out;
    float* ws  = (float*)d_ws;

    float* Y1  = ws;                       // 64  * 8192
    float* Y2  = ws + 64 * NCOL;           // 64  * 8192
    float* Y3  = ws + 128 * NCOL;          // 128 * 8192
    float* st1 = ws + 256 * NCOL;          // 128 floats
    float* st2 = st1 + 128;
    float* st3 = st2 + 128;

    // grouped_xyz: 8*1024*64*3 = 1,572,864 zeros = 393,216 float4
    k_zero<<<1536, 256, 0, stream>>>((float4*)out);

    k_gather_l1<<<32, 256, 0, stream>>>(points, onehot, idx, w1, Y1);
    k_stats<<<64, 256, 0, stream>>>(Y1, st1);

    // layer 2: 64 -> 64 : 4 M-tiles * 512 N-tiles = 2048 wave-tiles / 8 per block
    k_gemm_bn_relu_wmma<<<256, 256, 0, stream>>>(Y1, st1, g1, b1, w2, Y2, 64, 64);
    k_stats<<<64, 256, 0, stream>>>(Y2, st2);

    // layer 3: 64 -> 128 : 8 * 512 = 4096 wave-tiles / 8 per block
    k_gemm_bn_relu_wmma<<<512, 256, 0, stream>>>(Y2, st2, g2, b2, w3, Y3, 64, 128);
    k_stats<<<128, 256, 0, stream>>>(Y3, st3);

    // pooled output starts after the grouped_xyz zeros
    k_finalize<<<4096, 256, 0, stream>>>(Y3, st3, g3, b3, out + 1572864);
}